// ScaledDotProductAttention_33904471835159
// MI455X (gfx1250) — compile-verified
//
#include <hip/hip_runtime.h>

// ScaledDotProductAttention (no softmax, faithful to reference):
//   attn = mask ? (Q/8) @ K^T : -1e9      [B,H,S,S]  (written out, non-temporal)
//   out  = attn @ V                        [B,H,S,D]
// B=4 H=16 S=2048 D=64, all fp32. Fused: P-tile stays on chip between GEMMs.
// CDNA5 paths: v_wmma_f32_16x16x4_f32, global_load_async_to_lds_b128 (+ASYNCcnt),
// ds_load/store (packed 2addr), global_prefetch_b8, non-temporal attn stores.

typedef __attribute__((ext_vector_type(2))) float v2f;
typedef __attribute__((ext_vector_type(8))) float v8f;

#define B_ 4
#define H_ 16
#define S_ 2048
#define D_ 64
#define INV_T (1.0f / 8.0f)
#define MASK_FILL_ (-1e9f)

#define KB 16              // columns (keys) per tile
#define WAVES 8            // waves per workgroup
#define QROWS (WAVES * 16) // query rows per workgroup

// LDS strides (floats), padded for bank-conflict avoidance + alignment:
//  KSTR=68: rows 16B-aligned (async b128 copies), b64 reads spread banks by 4*l
//  VSTR/PSTR=18: even (8B-aligned b64 reads), 18l mod 64 distinct for l=0..15
#define KSTR 68
#define VSTR 18
#define PSTR 18

__global__ __launch_bounds__(256) void attn_nosm_kernel(
    const float* __restrict__ Q, const float* __restrict__ K,
    const float* __restrict__ V, const int* __restrict__ Msk,
    float* __restrict__ Out, float* __restrict__ Attn)
{
    __shared__ float ktile[16 * KSTR];          // K[kblock 16 x 64]
    __shared__ float vtile[64 * VSTR];          // V^T: vtile[d][k]
    __shared__ float pbuf[WAVES][16 * PSTR];    // per-wave P transpose scratch

    const int tid  = threadIdx.x;
    const int wave = tid >> 5;
    const int lane = tid & 31;
    const int l    = lane & 15;   // row/col within 16
    const int half = lane >> 4;   // 0: K=0,1 pair  1: K=2,3 pair

    const int bh    = blockIdx.y;          // 0..63
    const int b     = bh / H_;
    const int qbase = blockIdx.x * QROWS + wave * 16;
    const size_t bhS = (size_t)bh * S_;

    // ---- preload this wave's 16x64 Q block in WMMA-A fragment layout ----
    // A(16x4) layout: lanes 0-15 row M=l hold {K=4s,4s+1}; lanes 16-31 hold {4s+2,4s+3}
    v2f qa[16];
    {
        const float* qrow = Q + (bhS + (size_t)(qbase + l)) * D_;
        #pragma unroll
        for (int s = 0; s < 16; ++s) {
            v2f t = *(const v2f*)(qrow + 4 * s + 2 * half);
            qa[s] = t * INV_T;   // fold 1/temperature into Q
        }
    }

    v8f acc0 = {}, acc1 = {}, acc2 = {}, acc3 = {};   // out[16 x 64] accumulators

    // cooperative loader coords: 256 threads -> 16 rows x 16 float4 groups
    const int crow = tid >> 4;   // 0..15
    const int cg   = tid & 15;   // 0..15

    // LDS byte offset for this lane's K-tile slot (addr[31:0] == LDS offset)
    const unsigned klds = (unsigned)(size_t)(&ktile[crow * KSTR + cg * 4]);

    float* pb = pbuf[wave];

    for (int kb = 0; kb < S_ / KB; ++kb) {
        const int kbase = kb * KB;

        __syncthreads();   // previous tile fully consumed by all waves
        {
            const float* krow = K + (bhS + (size_t)(kbase + crow)) * D_ + cg * 4;
            const float* vrow = V + (bhS + (size_t)(kbase + crow)) * D_ + cg * 4;

            // K tile: async DMA global -> LDS, 16B per lane, no VGPR round trip
            asm volatile("global_load_async_to_lds_b128 %0, %1, off"
                         :: "v"(klds), "v"(krow) : "memory");

            // V tile: VGPR path (transposed on the way into LDS)
            float4 v4 = *(const float4*)vrow;
            vtile[(cg * 4 + 0) * VSTR + crow] = v4.x;
            vtile[(cg * 4 + 1) * VSTR + crow] = v4.y;
            vtile[(cg * 4 + 2) * VSTR + crow] = v4.z;
            vtile[(cg * 4 + 3) * VSTR + crow] = v4.w;

            if (kb + 1 < S_ / KB) {                     // global_prefetch_b8 next tile
                __builtin_prefetch(krow + KB * D_, 0, 0);
                __builtin_prefetch(vrow + KB * D_, 0, 0);
            }
            // this wave's async copies must land before it signals the barrier
            asm volatile("s_wait_asynccnt 0x0" ::: "memory");
        }
        __syncthreads();

        // ---- S-tile = (Q/8) @ K^T : 16 x v_wmma_f32_16x16x4_f32 over D=64 ----
        v8f c = {};
        #pragma unroll
        for (int s = 0; s < 16; ++s) {
            // B(4x16): lanes 0-15 col N=l hold {K=4s,4s+1}; lanes 16-31 {4s+2,4s+3}
            v2f bk = *(const v2f*)(&ktile[l * KSTR + 4 * s + 2 * half]);
            c = __builtin_amdgcn_wmma_f32_16x16x4_f32(
                    false, qa[s], false, bk, (short)0, c, false, false);
        }

        // ---- mask: column n = kbase + l is uniform across a lane's 8 values ----
        const int mv = Msk[(size_t)b * S_ + kbase + l];
        v8f cm;
        #pragma unroll
        for (int r = 0; r < 8; ++r) cm[r] = (mv == 0) ? MASK_FILL_ : c[r];

        // ---- write attn tile, non-temporal (never re-read) ----
        {
            float* arow = Attn + (bhS + (size_t)qbase) * S_ + kbase + l;
            #pragma unroll
            for (int r = 0; r < 8; ++r) {
                const int m = r + 8 * half;   // C/D layout: vgpr r -> row m
                __builtin_nontemporal_store(cm[r], arow + (size_t)m * S_);
            }
        }

        // ---- transpose P tile via wave-private LDS (C layout -> A layout) ----
        #pragma unroll
        for (int r = 0; r < 8; ++r) pb[(r + 8 * half) * PSTR + l] = cm[r];
        // compiler inserts s_wait_dscnt before the dependent ds_loads below

        // ---- out[16x64] += P(16x16) @ V(16x64): 16 WMMAs ----
        #pragma unroll
        for (int t = 0; t < 4; ++t) {
            v2f ap = *(const v2f*)(&pb[l * PSTR + 4 * t + 2 * half]);
            v2f b0 = *(const v2f*)(&vtile[( 0 + l) * VSTR + 4 * t + 2 * half]);
            v2f b1 = *(const v2f*)(&vtile[(16 + l) * VSTR + 4 * t + 2 * half]);
            v2f b2 = *(const v2f*)(&vtile[(32 + l) * VSTR + 4 * t + 2 * half]);
            v2f b3 = *(const v2f*)(&vtile[(48 + l) * VSTR + 4 * t + 2 * half]);
            acc0 = __builtin_amdgcn_wmma_f32_16x16x4_f32(false, ap, false, b0, (short)0, acc0, false, false);
            acc1 = __builtin_amdgcn_wmma_f32_16x16x4_f32(false, ap, false, b1, (short)0, acc1, false, false);
            acc2 = __builtin_amdgcn_wmma_f32_16x16x4_f32(false, ap, false, b2, (short)0, acc2, false, false);
            acc3 = __builtin_amdgcn_wmma_f32_16x16x4_f32(false, ap, false, b3, (short)0, acc3, false, false);
        }
    }

    // ---- write output block [16 x 64] ----
    {
        float* orow = Out + (bhS + (size_t)qbase) * D_ + l;
        #pragma unroll
        for (int r = 0; r < 8; ++r) {
            const int m = r + 8 * half;
            orow[(size_t)m * D_ +  0] = acc0[r];
            orow[(size_t)m * D_ + 16] = acc1[r];
            orow[(size_t)m * D_ + 32] = acc2[r];
            orow[(size_t)m * D_ + 48] = acc3[r];
        }
    }
}

extern "C" void kernel_launch(void* const* d_in, const int* in_sizes, int n_in,
                              void* d_out, int out_size, void* d_ws, size_t ws_size,
                              hipStream_t stream) {
    (void)in_sizes; (void)n_in; (void)out_size; (void)d_ws; (void)ws_size;
    const float* q = (const float*)d_in[0];
    const float* k = (const float*)d_in[1];
    const float* v = (const float*)d_in[2];
    const int*   m = (const int*)d_in[3];

    float* out  = (float*)d_out;                                   // [B,H,S,D]
    float* attn = out + (size_t)B_ * H_ * S_ * D_;                 // [B,H,S,S]

    dim3 grid(S_ / QROWS, B_ * H_);   // (16, 64)
    dim3 block(WAVES * 32);           // 256 threads = 8 waves (wave32)
    attn_nosm_kernel<<<grid, block, 0, stream>>>(q, k, v, m, out, attn);
}